// TemporalGNN_34222299415110
// MI455X (gfx1250) — compile-verified
//
#include <hip/hip_runtime.h>
#include <math.h>

typedef float v2f __attribute__((ext_vector_type(2)));
typedef float v8f __attribute__((ext_vector_type(8)));

#define FDIM 8
#define TP   12
#define FT   96      // FDIM * TP
#define UD   256

// Branch-free fast transcendentals (v_exp_f32 is 2^x; v_rcp_f32)
__device__ __forceinline__ float fast_exp(float x) {
    return __builtin_amdgcn_exp2f(x * 1.44269504088896341f);
}
__device__ __forceinline__ float fast_rcp(float x) {
    return __builtin_amdgcn_rcpf(x);
}

// ---------------------------------------------------------------------------
// Degree / normalization:  deg[i] = 1 (self loop) + sum_{e: col(e)=i} w[e]
// ---------------------------------------------------------------------------
__global__ void k_deg_init(float* __restrict__ deg, int n) {
    int i = blockIdx.x * blockDim.x + threadIdx.x;
    if (i < n) deg[i] = 1.0f;
}

__global__ void k_deg_scatter(const int* __restrict__ ei, const float* __restrict__ w,
                              float* __restrict__ deg, int e) {
    int i = blockIdx.x * blockDim.x + threadIdx.x;
    if (i < e) atomicAdd(&deg[ei[e + i]], w[i]);   // col = edge_index[1][i]
}

__global__ void k_dinv(float* __restrict__ deg, int n) {
    int i = blockIdx.x * blockDim.x + threadIdx.x;
    if (i < n) {
        float d = deg[i];
        deg[i] = (d > 0.0f) ? rsqrtf(d) : 0.0f;
    }
}

// ---------------------------------------------------------------------------
// Aggregation over all F*T channels at once:  Y = D^-1/2 (A + I) D^-1/2 X
// Y layout mirrors x: [node][f*TP + t]
// ---------------------------------------------------------------------------
__global__ void k_y_init(const float* __restrict__ x, const float* __restrict__ dinv,
                         float* __restrict__ Y, int n) {
    int i = blockIdx.x * blockDim.x + threadIdx.x;
    if (i < n * FT) {
        int node = i / FT;
        float di = dinv[node];
        Y[i] = di * di * x[i];
    }
}

__global__ void k_y_scatter(const float* __restrict__ x, const int* __restrict__ ei,
                            const float* __restrict__ w, const float* __restrict__ dinv,
                            float* __restrict__ Y, int e) {
    long long i = (long long)blockIdx.x * blockDim.x + threadIdx.x;
    long long tot = (long long)e * FT;
    if (i < tot) {
        int eid = (int)(i / FT);
        int c   = (int)(i % FT);
        int r   = ei[eid];       // source
        int cl  = ei[e + eid];   // destination
        float coeff = dinv[r] * w[eid] * dinv[cl];
        atomicAdd(&Y[(long long)cl * FT + c], coeff * x[(long long)r * FT + c]);
    }
}

// Transpose to period-major for vectorized A-fragment loads:
//   Y2[t][node][f] = Y[node][f*TP + t]
__global__ void k_y_t(const float* __restrict__ Y, float* __restrict__ Y2, int n) {
    int i = blockIdx.x * blockDim.x + threadIdx.x;   // over n*FT, dst-linear
    if (i < n * FT) {
        int f    = i & (FDIM - 1);
        int node = (i >> 3) % n;
        int t    = (i >> 3) / n;
        Y2[i] = Y[(long long)node * FT + f * TP + t];
    }
}

// ---------------------------------------------------------------------------
// Weight fusion:  M[f][u] = sum_k Wg[f][k] * L[k][u]   (top U rows of L only)
//                 c[u]    = sum_k bg[k]   * L[k][u] + Lb[u]
// ---------------------------------------------------------------------------
__global__ void k_fuse_w(const float* __restrict__ Wg, const float* __restrict__ L,
                         float* __restrict__ M) {
    int i = blockIdx.x * blockDim.x + threadIdx.x;
    if (i < FDIM * UD) {
        int f = i / UD, u = i % UD;
        float acc = 0.0f;
        for (int k = 0; k < UD; ++k)
            acc += Wg[f * UD + k] * L[k * UD + u];
        M[i] = acc;
    }
}

__global__ void k_fuse_b(const float* __restrict__ bg, const float* __restrict__ L,
                         const float* __restrict__ Lb, float* __restrict__ c) {
    int u = blockIdx.x * blockDim.x + threadIdx.x;
    if (u < UD) {
        float acc = Lb[u];
        for (int k = 0; k < UD; ++k)
            acc += bg[k] * L[k * UD + u];
        c[u] = acc;
    }
}

__global__ void k_softmax(const float* __restrict__ att, float* __restrict__ probs) {
    float mx = att[0];
    for (int t = 1; t < TP; ++t) mx = fmaxf(mx, att[t]);
    float ex[TP];
    float s = 0.0f;
    for (int t = 0; t < TP; ++t) { ex[t] = fast_exp(att[t] - mx); s += ex[t]; }
    float rs = 1.0f / s;
    for (int t = 0; t < TP; ++t) probs[t] = ex[t] * rs;
}

// ---------------------------------------------------------------------------
// Main TGCN kernel: one wave owns a (16-node x 16-unit) tile, loops T periods.
//   zc = Y_t @ Mz + cz ;  hc = Y_t @ Mh + ch
//   H += p_t * sigmoid(-zc) * tanh(hc)       [since (1-sigmoid(z)) = sigmoid(-z)]
// K = 8 done as two V_WMMA_F32_16X16X4_F32 per gate.
// ---------------------------------------------------------------------------
__global__ void k_tgcn_wmma(const float* __restrict__ Y2,
                            const float* __restrict__ Mz, const float* __restrict__ Mh,
                            const float* __restrict__ cz, const float* __restrict__ ch,
                            const float* __restrict__ probs,
                            float* __restrict__ H, int n) {
    int wid  = (blockIdx.x * blockDim.x + threadIdx.x) >> 5;  // wave-uniform
    int lane = threadIdx.x & 31;
    int u_tile = wid & 15;
    int n_tile = wid >> 4;
    int nbase  = n_tile * 16;
    if (nbase >= n) return;                 // wave-uniform exit: EXEC stays all-ones

    int m     = lane & 15;                  // row (A) / column (B,C) index within tile
    int khalf = lane >> 4;                  // lane-half selects K pairs {0,1} vs {2,3}
    int col   = u_tile * 16 + m;

    // B fragments (period-invariant). Layout: VGPRv -> K = v + 2*khalf, N = lane&15.
    v2f bz0, bz1, bh0, bh1;
    bz0.x = Mz[(2 * khalf + 0) * UD + col];
    bz0.y = Mz[(2 * khalf + 1) * UD + col];
    bz1.x = Mz[(4 + 2 * khalf) * UD + col];
    bz1.y = Mz[(5 + 2 * khalf) * UD + col];
    bh0.x = Mh[(2 * khalf + 0) * UD + col];
    bh0.y = Mh[(2 * khalf + 1) * UD + col];
    bh1.x = Mh[(4 + 2 * khalf) * UD + col];
    bh1.y = Mh[(5 + 2 * khalf) * UD + col];

    float czv = cz[col];
    float chv = ch[col];

    v8f hacc;
    #pragma unroll
    for (int v = 0; v < 8; ++v) hacc[v] = 0.0f;

    int node_a = nbase + m;                 // A-fragment row (clamped for partial tiles)
    if (node_a >= n) node_a = n - 1;

    for (int t = 0; t < TP; ++t) {
        float p = probs[t];

        // A fragments: contiguous b64 loads from period-major Y2[t][node][0..7]
        const float* arow = Y2 + ((long long)t * n + node_a) * FDIM;
        v2f a0 = *(const v2f*)(arow + 2 * khalf);        // K = 2*khalf + {0,1}
        v2f a1 = *(const v2f*)(arow + 4 + 2 * khalf);    // K = 4 + 2*khalf + {0,1}

        v8f zc, hc;
        #pragma unroll
        for (int v = 0; v < 8; ++v) { zc[v] = czv; hc[v] = chv; }

        zc = __builtin_amdgcn_wmma_f32_16x16x4_f32(false, a0, false, bz0, (short)0, zc, false, false);
        zc = __builtin_amdgcn_wmma_f32_16x16x4_f32(false, a1, false, bz1, (short)0, zc, false, false);
        hc = __builtin_amdgcn_wmma_f32_16x16x4_f32(false, a0, false, bh0, (short)0, hc, false, false);
        hc = __builtin_amdgcn_wmma_f32_16x16x4_f32(false, a1, false, bh1, (short)0, hc, false, false);

        #pragma unroll
        for (int v = 0; v < 8; ++v) {
            // (1 - sigmoid(zc)) = rcp(1 + exp(zc));  saturates to 0/1 correctly
            float oz = fast_rcp(1.0f + fast_exp(zc[v]));
            // tanh(hc) = 1 - 2*rcp(exp(2*hc) + 1);   saturates to +/-1 correctly
            float th = 1.0f - 2.0f * fast_rcp(1.0f + fast_exp(2.0f * hc[v]));
            hacc[v] += p * oz * th;
        }
    }

    // C/D layout: VGPR v -> M = v + 8*khalf, N = lane&15
    #pragma unroll
    for (int v = 0; v < 8; ++v) {
        int node = nbase + v + 8 * khalf;
        if (node < n) H[(long long)node * UD + col] = hacc[v];
    }
}

// ---------------------------------------------------------------------------
// Final:  out[n,t] = relu(H[n,:]) @ lin_w[:,t] + lin_b[t]
// ---------------------------------------------------------------------------
__global__ void k_out(const float* __restrict__ H, const float* __restrict__ lin_w,
                      const float* __restrict__ lin_b, float* __restrict__ out, int n) {
    int i = blockIdx.x * blockDim.x + threadIdx.x;
    if (i < n * TP) {
        int node = i / TP, t = i % TP;
        const float* hrow = H + (long long)node * UD;
        float acc = lin_b[t];
        for (int u = 0; u < UD; ++u)
            acc += fmaxf(hrow[u], 0.0f) * lin_w[u * TP + t];
        out[i] = acc;
    }
}

// ---------------------------------------------------------------------------
extern "C" void kernel_launch(void* const* d_in, const int* in_sizes, int n_in,
                              void* d_out, int out_size, void* d_ws, size_t ws_size,
                              hipStream_t stream) {
    const float* x     = (const float*)d_in[0];
    const int*   ei    = (const int*)  d_in[1];
    const float* ew    = (const float*)d_in[2];
    const float* att   = (const float*)d_in[3];
    const float* Wz    = (const float*)d_in[4];
    const float* bz    = (const float*)d_in[5];
    // d_in[6]/[7] (Wr/br) and d_in[12]/[13] (Lrw/Lrb): dead code since H0 == 0
    const float* Wh    = (const float*)d_in[8];
    const float* bh    = (const float*)d_in[9];
    const float* Lzw   = (const float*)d_in[10];
    const float* Lzb   = (const float*)d_in[11];
    const float* Lhw   = (const float*)d_in[14];
    const float* Lhb   = (const float*)d_in[15];
    const float* lin_w = (const float*)d_in[16];
    const float* lin_b = (const float*)d_in[17];

    int n = in_sizes[0] / FT;     // 20000
    int e = in_sizes[2];          // 320000

    float* ws   = (float*)d_ws;
    size_t o = 0;
    float* dinv  = ws + o;  o += (size_t)n;
    float* Y     = ws + o;  o += (size_t)n * FT;
    float* Y2    = ws + o;  o += (size_t)n * FT;
    float* Mz    = ws + o;  o += FDIM * UD;
    float* Mh    = ws + o;  o += FDIM * UD;
    float* czp   = ws + o;  o += UD;
    float* chp   = ws + o;  o += UD;
    float* probs = ws + o;  o += 16;
    float* H     = ws + o;  o += (size_t)n * UD;

    const int B = 256;
    k_deg_init   <<<(n + B - 1) / B, B, 0, stream>>>(dinv, n);
    k_deg_scatter<<<(e + B - 1) / B, B, 0, stream>>>(ei, ew, dinv, e);
    k_dinv       <<<(n + B - 1) / B, B, 0, stream>>>(dinv, n);
    k_y_init     <<<(n * FT + B - 1) / B, B, 0, stream>>>(x, dinv, Y, n);
    long long tot = (long long)e * FT;
    k_y_scatter  <<<(int)((tot + B - 1) / B), B, 0, stream>>>(x, ei, ew, dinv, Y, e);
    k_y_t        <<<(n * FT + B - 1) / B, B, 0, stream>>>(Y, Y2, n);
    k_fuse_w     <<<(FDIM * UD + B - 1) / B, B, 0, stream>>>(Wz, Lzw, Mz);
    k_fuse_w     <<<(FDIM * UD + B - 1) / B, B, 0, stream>>>(Wh, Lhw, Mh);
    k_fuse_b     <<<(UD + B - 1) / B, B, 0, stream>>>(bz, Lzw, Lzb, czp);
    k_fuse_b     <<<(UD + B - 1) / B, B, 0, stream>>>(bh, Lhw, Lhb, chp);
    k_softmax    <<<1, 1, 0, stream>>>(att, probs);

    int ntiles  = (n + 15) / 16;          // 1250 exact tiles for N=20000
    int threads = ntiles * 16 * 32;       // 16 u-tiles per node-tile, wave32 each
    k_tgcn_wmma  <<<threads / 256, 256, 0, stream>>>(Y2, Mz, Mh, czp, chp, probs, H, n);

    k_out        <<<(n * TP + B - 1) / B, B, 0, stream>>>(H, lin_w, lin_b, (float*)d_out, n);
}